// Structure2VecLayer_88399016886795
// MI455X (gfx1250) — compile-verified
//
#include <hip/hip_runtime.h>
#include <hip/hip_bf16.h>

typedef float v2f __attribute__((ext_vector_type(2)));
typedef float v8f __attribute__((ext_vector_type(8)));

#define NN 50000
#define EE 800000
#define HH 256
#define FB 128
#define EPSV 1e-5f

// ---- workspace layout (float offsets) ----
#define OFF_AGG  0u                      // [N,128] segment_sum(edge_w)
#define OFF_H1   6400000u                // [N,256] h1, reused in-place as r1
#define OFF_DEG  19200000u               // [N] degree (float)
#define OFF_S1   19250000u               // [512] stats layer1 (sum, sumsq)
#define OFF_S2   19250512u               // [512] stats layer2
#define ZERO_TOT 19251024u               // floats to zero each launch
#define OFF_SC1  19251024u               // [256] BN1 scale
#define OFF_SH1  19251280u               // [256] BN1 shift
#define OFF_W2P  19251536u               // [256*256] scale-folded W2
#define OFF_C2   19317072u               // [256] folded column bias for layer 2

// CDNA5 async global->LDS copy path (ASYNCcnt), with synchronous fallback.
#if defined(__AMDGCN__) && __has_builtin(__builtin_amdgcn_global_load_async_to_lds_b128) && __has_builtin(__builtin_amdgcn_s_wait_asynccnt)
#define USE_ASYNC_LDS 1
#else
#define USE_ASYNC_LDS 0
#endif

#if USE_ASYNC_LDS
typedef int v4i_gcc __attribute__((vector_size(16)));
typedef __attribute__((address_space(1))) v4i_gcc* g_v4i_ptr;  // global
typedef __attribute__((address_space(3))) v4i_gcc* l_v4i_ptr;  // LDS

__device__ __forceinline__ void async_g2l_b128(const void* g, const void* l) {
  // generic->AS1 is value-preserving; generic LDS pointer's low 32 bits are the
  // LDS byte offset (ISA flat-aperture rule), which is the AS3 address.
  __builtin_amdgcn_global_load_async_to_lds_b128(
      (g_v4i_ptr)(unsigned long long)g,
      (l_v4i_ptr)(unsigned int)(unsigned long long)l,
      0, 0);
}
#endif

// ---------------- zero accumulators ----------------
__global__ void zero_ws(float* ws, unsigned n) {
  unsigned i = blockIdx.x * blockDim.x + threadIdx.x;
  unsigned stride = gridDim.x * blockDim.x;
  for (; i < n; i += stride) ws[i] = 0.f;
}

// ---------------- edge scatter: one wave (32 lanes) per edge ----------------
__global__ __launch_bounds__(256) void edge_scatter(
    const float* __restrict__ feat, const float* __restrict__ ew,
    const int* __restrict__ src, const int* __restrict__ dst,
    float* __restrict__ h1, float* __restrict__ agg, float* __restrict__ deg) {
  int gw = (blockIdx.x * blockDim.x + threadIdx.x) >> 5;
  int lane = threadIdx.x & 31;
  if (gw >= EE) return;
  int s = src[gw];
  int d = dst[gw];
  const float* fs = feat + (size_t)s * HH;
  float* hd = h1 + (size_t)d * HH;
#pragma unroll
  for (int j = lane; j < HH; j += 32) atomicAdd(&hd[j], fs[j]);
  const float* es = ew + (size_t)gw * FB;
  float* ad = agg + (size_t)d * FB;
#pragma unroll
  for (int j = lane; j < FB; j += 32) atomicAdd(&ad[j], es[j]);
  if (lane == 0) atomicAdd(&deg[d], 1.0f);
}

// ---------------- fused WMMA GEMM + bias + ReLU + column stats ----------------
// C[Nrows,256] = [A1|A2] @ [B1;B2] + biasCol + deg*degBiasCol + resid, ReLU,
// store, accumulate per-column sum/sumsq into stats[0..255]/[256..511].
#define BM 32
#define BK 32
#define A_STR 36    // row-major A tile pitch: 36*4B = 144B (16B-aligned rows);
                    // lane banks {36*i mod 64} distinct; k-groups (+2) disjoint
#define B_STR 264   // k-major B tile pitch: 2*264 % 64 == 16 -> the two 16-lane
                    // k-groups read disjoint LDS bank windows; 264*4B 16B-aligned

__global__ __launch_bounds__(256) void gemm_relu_stats(
    const float* __restrict__ A1, int K1,
    const float* __restrict__ A2, int K2,
    const float* __restrict__ B1, const float* __restrict__ B2,
    const float* __restrict__ biasCol,
    const float* __restrict__ degRow,       // nullable
    const float* __restrict__ degBiasCol,   // nullable
    const float* __restrict__ resid,        // nullable
    float* __restrict__ Cout,
    float* __restrict__ stats,
    int Nrows) {
  __shared__ __align__(16) float Asl[BM * A_STR];   // row-major A tile [m][k]
  __shared__ __align__(16) float Bsl[BK * B_STR];   // k-major B tile [k][n]
  __shared__ float sSum[256];
  __shared__ float sSq[256];

  const int tid    = threadIdx.x;
  const int lane   = tid & 31;
  const int wave   = tid >> 5;
  const int lanelo = lane & 15;
  const int hi     = lane >> 4;       // which 16-lane half
  const int koff   = hi * 2;          // A/B fragment k-offset per ISA layout
  const int wrow   = (wave >> 2) * 16;
  const int wcol   = (wave & 3) * 64;
  const int mbase  = blockIdx.x * BM;
  const int Ktot   = K1 + K2;

  if (tid < 256) { sSum[tid] = 0.f; sSq[tid] = 0.f; }

  v8f acc0 = {}, acc1 = {}, acc2 = {}, acc3 = {};

  for (int kb = 0; kb < Ktot; kb += BK) {
    __syncthreads();
    // ---- A tile (BM x BK): one float4 per thread, row-major
    {
      const int r  = tid >> 3;
      const int c4 = (tid & 7) * 4;
      const float* Asrc;
      int Kw, scol;
      if (kb < K1) { Asrc = A1; Kw = K1; scol = kb; }
      else         { Asrc = A2; Kw = K2; scol = kb - K1; }
      int rowg = mbase + r;
      if (rowg >= Nrows) rowg = Nrows - 1;  // clamp: values only feed unstored rows
      const float* gsrc = Asrc + (size_t)rowg * Kw + scol + c4;
      float* ldst = Asl + r * A_STR + c4;
#if USE_ASYNC_LDS
      async_g2l_b128(gsrc, ldst);
#else
      *(float4*)ldst = *(const float4*)gsrc;
#endif
    }
    // ---- B tile (BK x 256): eight float4 per thread, k-major
    {
      const float* Bsrc;
      int srow;
      if (kb < K1) { Bsrc = B1; srow = kb; }
      else         { Bsrc = B2; srow = kb - K1; }
#pragma unroll
      for (int t = 0; t < 8; ++t) {
        int i4 = tid + t * 256;
        int r  = i4 >> 6;
        int c  = (i4 & 63) << 2;
        const float* gsrc = Bsrc + (size_t)(srow + r) * HH + c;
        float* ldst = Bsl + r * B_STR + c;
#if USE_ASYNC_LDS
        async_g2l_b128(gsrc, ldst);
#else
        *(float4*)ldst = *(const float4*)gsrc;
#endif
      }
    }
#if USE_ASYNC_LDS
    __builtin_amdgcn_s_wait_asynccnt(0);
#endif
    __syncthreads();
    // ---- compute: 8 k-steps of V_WMMA_F32_16X16X4_F32 x 4 n-subtiles
#pragma unroll
    for (int kk = 0; kk < BK; kk += 4) {
      v2f a = *(const v2f*)(Asl + (wrow + lanelo) * A_STR + kk + koff);
      const float* Bp0 = Bsl + (kk + koff) * B_STR + wcol + lanelo;
      const float* Bp1 = Bp0 + B_STR;
      v2f b0, b1v, b2v, b3v;
      b0.x  = Bp0[0];  b0.y  = Bp1[0];
      b1v.x = Bp0[16]; b1v.y = Bp1[16];
      b2v.x = Bp0[32]; b2v.y = Bp1[32];
      b3v.x = Bp0[48]; b3v.y = Bp1[48];
      acc0 = __builtin_amdgcn_wmma_f32_16x16x4_f32(false, a, false, b0,  (short)0, acc0, false, false);
      acc1 = __builtin_amdgcn_wmma_f32_16x16x4_f32(false, a, false, b1v, (short)0, acc1, false, false);
      acc2 = __builtin_amdgcn_wmma_f32_16x16x4_f32(false, a, false, b2v, (short)0, acc2, false, false);
      acc3 = __builtin_amdgcn_wmma_f32_16x16x4_f32(false, a, false, b3v, (short)0, acc3, false, false);
    }
  }
  __syncthreads();

  // ---- epilogue: bias + ReLU + store + LDS-reduced column stats
  v8f accs[4] = {acc0, acc1, acc2, acc3};
#pragma unroll
  for (int t = 0; t < 4; ++t) {
    int col = wcol + t * 16 + lanelo;
    float bcol = biasCol[col];
    float dbc  = degBiasCol ? degBiasCol[col] : 0.f;
    float s = 0.f, q = 0.f;
#pragma unroll
    for (int v = 0; v < 8; ++v) {
      int row = mbase + wrow + v + 8 * hi;   // C-fragment layout: M = v (+8 for hi half)
      if (row < Nrows) {
        float val = accs[t][v] + bcol;
        if (degRow) val += degRow[row] * dbc;
        if (resid)  val += resid[(size_t)row * HH + col];
        val = fmaxf(val, 0.f);
        Cout[(size_t)row * HH + col] = val;
        s += val;
        q += val * val;
      }
    }
    atomicAdd(&sSum[col], s);
    atomicAdd(&sSq[col], q);
  }
  __syncthreads();
  if (tid < 256) {
    atomicAdd(&stats[tid],       sSum[tid]);
    atomicAdd(&stats[256 + tid], sSq[tid]);
  }
}

// ---------------- BN fold: scale/shift from stats ----------------
__global__ void bn_fold(const float* __restrict__ stats,
                        const float* __restrict__ gamma, const float* __restrict__ beta,
                        float* __restrict__ scale, float* __restrict__ shift) {
  int j = threadIdx.x;
  float mean = stats[j] / (float)NN;
  float var  = stats[256 + j] / (float)NN - mean * mean;
  float inv  = rsqrtf(var + EPSV);
  float sc   = gamma[j] * inv;
  scale[j] = sc;
  shift[j] = beta[j] - mean * sc;
}

// W2'[k,j] = scale1[k] * W2[k,j]
__global__ void make_w2p(const float* __restrict__ W2, const float* __restrict__ scale,
                         float* __restrict__ W2p) {
  int k = blockIdx.x;
  int j = threadIdx.x;
  W2p[k * HH + j] = scale[k] * W2[k * HH + j];
}

// c2[j] = b2[j] + sum_k shift1[k] * W2[k,j]
__global__ void make_c2(const float* __restrict__ W2, const float* __restrict__ shift,
                        const float* __restrict__ b2, float* __restrict__ c2) {
  int j = threadIdx.x;
  float acc = b2[j];
  for (int k = 0; k < HH; ++k) acc += shift[k] * W2[k * HH + j];
  c2[j] = acc;
}

// ---------------- BN2 apply in place on d_out ----------------
__global__ void bn2_apply(float* __restrict__ out, const float* __restrict__ stats,
                          const float* __restrict__ gamma, const float* __restrict__ beta) {
  size_t i = (size_t)blockIdx.x * blockDim.x + threadIdx.x;
  size_t stride = (size_t)gridDim.x * blockDim.x;
  size_t total = (size_t)NN * HH;
  for (; i < total; i += stride) {
    int j = (int)(i & (HH - 1));
    float mean = stats[j] / (float)NN;
    float var  = stats[256 + j] / (float)NN - mean * mean;
    out[i] = (out[i] - mean) * rsqrtf(var + EPSV) * gamma[j] + beta[j];
  }
}

extern "C" void kernel_launch(void* const* d_in, const int* in_sizes, int n_in,
                              void* d_out, int out_size, void* d_ws, size_t ws_size,
                              hipStream_t stream) {
  (void)in_sizes; (void)n_in; (void)out_size; (void)ws_size;
  const float* features = (const float*)d_in[0];
  const float* edge_w   = (const float*)d_in[1];
  const int*   src      = (const int*)d_in[2];
  const int*   dst      = (const int*)d_in[3];
  const float* bond_W   = (const float*)d_in[4];
  const float* bond_b   = (const float*)d_in[5];
  const float* W1       = (const float*)d_in[6];
  const float* b1       = (const float*)d_in[7];
  const float* W2       = (const float*)d_in[8];
  const float* b2       = (const float*)d_in[9];
  const float* bn1_g    = (const float*)d_in[10];
  const float* bn1_b    = (const float*)d_in[11];
  const float* bn2_g    = (const float*)d_in[12];
  const float* bn2_b    = (const float*)d_in[13];
  float* out = (float*)d_out;

  float* ws     = (float*)d_ws;
  float* agg    = ws + OFF_AGG;
  float* h1     = ws + OFF_H1;    // reused in place as r1
  float* deg    = ws + OFF_DEG;
  float* stats1 = ws + OFF_S1;
  float* stats2 = ws + OFF_S2;
  float* scale1 = ws + OFF_SC1;
  float* shift1 = ws + OFF_SH1;
  float* w2p    = ws + OFF_W2P;
  float* c2     = ws + OFF_C2;

  // 1. zero accumulators
  zero_ws<<<(ZERO_TOT + 1023) / 1024, 1024, 0, stream>>>(ws, ZERO_TOT);

  // 2. edge scatter: h1 += features[src], agg += edge_w, deg += 1 (onto dst)
  edge_scatter<<<EE / 8, 256, 0, stream>>>(features, edge_w, src, dst, h1, agg, deg);

  // 3. layer 1: relu([h1|agg] @ [W1;bond_W] + b1 + deg*bond_b) -> h1 (in place), stats1
  int gblocks = (NN + BM - 1) / BM;
  gemm_relu_stats<<<gblocks, 256, 0, stream>>>(
      h1, HH, agg, FB, W1, bond_W, b1, deg, bond_b, nullptr, h1, stats1, NN);

  // 4. fold BN1 into layer-2 weights
  bn_fold<<<1, 256, 0, stream>>>(stats1, bn1_g, bn1_b, scale1, shift1);
  make_w2p<<<HH, HH, 0, stream>>>(W2, scale1, w2p);
  make_c2<<<1, HH, 0, stream>>>(W2, shift1, b2, c2);

  // 5. layer 2: relu(r1 @ W2' + c2 + features) -> d_out, stats2
  gemm_relu_stats<<<gblocks, 256, 0, stream>>>(
      h1, HH, nullptr, 0, w2p, nullptr, c2, nullptr, nullptr, features, out, stats2, NN);

  // 6. BN2 in place on d_out
  bn2_apply<<<4096, 256, 0, stream>>>(out, stats2, bn2_g, bn2_b);
}